// CAM_68453188763683
// MI455X (gfx1250) — compile-verified
//
#include <hip/hip_runtime.h>
#include <math.h>
#include <float.h>

typedef float v2f __attribute__((ext_vector_type(2)));
typedef float v8f __attribute__((ext_vector_type(8)));

#define LB    96          // L*B = 3*32
#define C     256
#define P49   49
#define Q961  961
#define NCH   8           // c-chunks for xf norm partials
#define TEMP_INV 40.0f    // 1/0.025
#define BN_EPS 1e-5f

// ---------------------------------------------------------------------------
// Kernel 1a: inverse channel norms for zf (small: 4.8 MB total).
// ---------------------------------------------------------------------------
__global__ void cam_norm(const float* __restrict__ f, float* __restrict__ inv, int PQ) {
    int lb = blockIdx.x;
    const float* fp = f + (size_t)lb * C * PQ;
    for (int p = threadIdx.x; p < PQ; p += blockDim.x) {
        float s = 0.f;
        for (int c = 0; c < C; ++c) { float v = fp[c * PQ + p]; s += v * v; }
        inv[(size_t)lb * PQ + p] = 1.0f / fmaxf(sqrtf(s), 1e-12f);
    }
}

// ---------------------------------------------------------------------------
// Kernel 1b: xf norm partial sums of squares over a 32-channel chunk.
// grid (96, 8), block 128.  partial[(chunk*LB + lb)*Q961 + q]
// ---------------------------------------------------------------------------
__global__ __launch_bounds__(128) void cam_norm2_partial(
    const float* __restrict__ xf, float* __restrict__ partial) {
    int lb = blockIdx.x, ch = blockIdx.y;
    const float* fp = xf + (size_t)lb * C * Q961 + (size_t)ch * (C / NCH) * Q961;
    float* out = partial + ((size_t)ch * LB + lb) * Q961;
    for (int q = threadIdx.x; q < Q961; q += 128) {
        float s = 0.f;
        #pragma unroll 4
        for (int c = 0; c < C / NCH; ++c) { float v = fp[c * Q961 + q]; s += v * v; }
        out[q] = s;
    }
}

// ---------------------------------------------------------------------------
// Kernel 1c: reduce partials -> inv2.  grid 96, block 256.
// ---------------------------------------------------------------------------
__global__ void cam_norm2_finish(const float* __restrict__ partial,
                                 float* __restrict__ inv2) {
    int lb = blockIdx.x;
    for (int q = threadIdx.x; q < Q961; q += blockDim.x) {
        float s = 0.f;
        #pragma unroll
        for (int ch = 0; ch < NCH; ++ch)
            s += partial[((size_t)ch * LB + lb) * Q961 + q];
        inv2[(size_t)lb * Q961 + q] = 1.0f / fmaxf(sqrtf(s), 1e-12f);
    }
}

// ---------------------------------------------------------------------------
// Kernel 2: A[lb][p][q] = inv1[p]*inv2[q] * sum_c f1[c,p]*f2[c,q]
// Raw fp32 WMMA 16x16x4 accumulation; normalization applied in the epilogue
// (separable diagonal scaling), so the inner loop is loads + wmma only.
// wave = 16-row M tile (4 waves cover 49 rows padded to 64),
// grid = (96 pairs, 61 N-tiles of 16).
// ---------------------------------------------------------------------------
__global__ __launch_bounds__(128) void cam_gemm(
    const float* __restrict__ zf, const float* __restrict__ xf,
    const float* __restrict__ inv1, const float* __restrict__ inv2,
    float* __restrict__ A) {
    int lb   = blockIdx.x;
    int nt   = blockIdx.y;                 // N tile (16 cols)
    int lane = threadIdx.x & 31;
    int wave = threadIdx.x >> 5;           // M tile (16 rows)

    const float* f1 = zf + (size_t)lb * C * P49;
    const float* f2 = xf + (size_t)lb * C * Q961;

    int mrow  = wave * 16 + (lane & 15);
    int ncol  = nt * 16 + (lane & 15);
    int khalf = (lane >> 4) * 2;           // 0 or 2

    // clamp padding lanes in-bounds; their results are never stored
    int am = (mrow < P49)  ? mrow : P49 - 1;
    int an = (ncol < Q961) ? ncol : Q961 - 1;

    v8f acc = {};
    #pragma unroll 4
    for (int k0 = 0; k0 < C; k0 += 4) {
        int k = k0 + khalf;
        v2f a, b;
        a.x = f1[(size_t)k       * P49  + am];
        a.y = f1[(size_t)(k + 1) * P49  + am];
        b.x = f2[(size_t)k       * Q961 + an];
        b.y = f2[(size_t)(k + 1) * Q961 + an];
        acc = __builtin_amdgcn_wmma_f32_16x16x4_f32(
            /*neg_a=*/false, a, /*neg_b=*/false, b,
            /*c_mod=*/(short)0, acc, /*reuse_a=*/false, /*reuse_b=*/false);
    }

    float* Ap = A + (size_t)lb * P49 * Q961;
    int n = nt * 16 + (lane & 15);
    if (n < Q961) {
        float sb = inv2[(size_t)lb * Q961 + n];
        int mbase = wave * 16 + (lane >> 4) * 8;
        #pragma unroll
        for (int r = 0; r < 8; ++r) {
            int m = mbase + r;
            if (m < P49)
                Ap[(size_t)m * Q961 + n] = acc[r] * inv1[(size_t)lb * P49 + m] * sb;
        }
    }
}

// ---------------------------------------------------------------------------
// Kernel 3: both attention heads for one (l,b) pair per block (1024 threads).
// ---------------------------------------------------------------------------
__global__ __launch_bounds__(1024) void cam_attn(
    const float* __restrict__ A_,
    const float* __restrict__ W1, const float* __restrict__ b1,
    const float* __restrict__ g1, const float* __restrict__ be1,
    const float* __restrict__ rm1, const float* __restrict__ rv1,
    const float* __restrict__ W2, const float* __restrict__ b2,
    const float* __restrict__ W3, const float* __restrict__ b3,
    const float* __restrict__ g3, const float* __restrict__ be3,
    const float* __restrict__ rm3, const float* __restrict__ rv3,
    const float* __restrict__ W4, const float* __restrict__ b4,
    float* __restrict__ att1, float* __restrict__ att2) {
    __shared__ float mq[Q961];     // mean over P
    __shared__ float tqs[Q961];    // conv2 output
    __shared__ float m2[64];       // mean over Q
    __shared__ float t1s[32];
    __shared__ float t3s[8];
    __shared__ float tps[64];
    __shared__ float s49[64];
    __shared__ float red[32];
    __shared__ float bcast[2];

    int lb = blockIdx.x;
    const float* A = A_ + (size_t)lb * P49 * Q961;
    int tid = threadIdx.x, lane = tid & 31, wave = tid >> 5;

    // mq[q] = mean_p A[p,q]
    for (int q = tid; q < Q961; q += 1024) {
        float s = 0.f;
        for (int p = 0; p < P49; ++p) s += A[(size_t)p * Q961 + q];
        mq[q] = s * (1.0f / 49.0f);
    }
    // m2[p] = mean_q A[p,q]  (wave-per-row, lane-strided over q)
    for (int p = wave; p < P49; p += 32) {
        float s = 0.f;
        for (int q = lane; q < Q961; q += 32) s += A[(size_t)p * Q961 + q];
        for (int off = 16; off > 0; off >>= 1) s += __shfl_down(s, off, 32);
        if (lane == 0) m2[p] = s * (1.0f / 961.0f);
    }
    __syncthreads();

    // bottleneck layers + BN + ReLU
    if (wave == 0 && lane < 31) {
        float s = b1[lane];
        for (int q = 0; q < Q961; ++q) s += mq[q] * W1[lane * Q961 + q];
        s = (s - rm1[lane]) * rsqrtf(rv1[lane] + BN_EPS) * g1[lane] + be1[lane];
        t1s[lane] = fmaxf(s, 0.f);
    }
    if (wave == 1 && lane < 7) {
        float s = b3[lane];
        for (int p = 0; p < P49; ++p) s += m2[p] * W3[lane * P49 + p];
        s = (s - rm3[lane]) * rsqrtf(rv3[lane] + BN_EPS) * g3[lane] + be3[lane];
        t3s[lane] = fmaxf(s, 0.f);
    }
    __syncthreads();

    // expand back: tq[q] (31 -> 961), tp[p] (7 -> 49)
    for (int q = tid; q < Q961; q += 1024) {
        float s = b2[q];
        for (int k = 0; k < 31; ++k) s += t1s[k] * W2[q * 31 + k];
        tqs[q] = s;
    }
    if (tid < P49) {
        float s = b4[tid];
        for (int k = 0; k < 7; ++k) s += t3s[k] * W4[tid * 7 + k];
        tps[tid] = s;
    }
    __syncthreads();

    // s49[p] = mean_q A[p,q]*tq[q]
    for (int p = wave; p < P49; p += 32) {
        float s = 0.f;
        for (int q = lane; q < Q961; q += 32) s += A[(size_t)p * Q961 + q] * tqs[q];
        for (int off = 16; off > 0; off >>= 1) s += __shfl_down(s, off, 32);
        if (lane == 0) s49[p] = s * (1.0f / 961.0f);
    }
    __syncthreads();

    // att1 = softmax(s49/T) + 1   (single wave handles the 49 values)
    if (wave == 0) {
        float v0 = (lane < P49)      ? s49[lane]      * TEMP_INV : -FLT_MAX;
        float v1 = (lane + 32 < P49) ? s49[lane + 32] * TEMP_INV : -FLT_MAX;
        float mx = fmaxf(v0, v1);
        for (int off = 16; off > 0; off >>= 1) mx = fmaxf(mx, __shfl_xor(mx, off, 32));
        float e0 = (lane < P49)      ? expf(v0 - mx) : 0.f;
        float e1 = (lane + 32 < P49) ? expf(v1 - mx) : 0.f;
        float ss = e0 + e1;
        for (int off = 16; off > 0; off >>= 1) ss += __shfl_xor(ss, off, 32);
        float rs = 1.0f / ss;
        if (lane < P49)      att1[(size_t)lb * P49 + lane]      = e0 * rs + 1.0f;
        if (lane + 32 < P49) att1[(size_t)lb * P49 + lane + 32] = e1 * rs + 1.0f;
    }

    // att2 = softmax over 961 of mean_p A[p,q]*tp[p] / T, block-wide
    float myz = -FLT_MAX;
    if (tid < Q961) {
        float s = 0.f;
        for (int p = 0; p < P49; ++p) s += A[(size_t)p * Q961 + tid] * tps[p];
        myz = s * (1.0f / 49.0f) * TEMP_INV;
    }
    float mx = myz;
    for (int off = 16; off > 0; off >>= 1) mx = fmaxf(mx, __shfl_xor(mx, off, 32));
    if (lane == 0) red[wave] = mx;
    __syncthreads();
    if (wave == 0) {
        float v = red[lane];
        for (int off = 16; off > 0; off >>= 1) v = fmaxf(v, __shfl_xor(v, off, 32));
        if (lane == 0) bcast[0] = v;
    }
    __syncthreads();
    float e = (tid < Q961) ? expf(myz - bcast[0]) : 0.f;
    float ss = e;
    for (int off = 16; off > 0; off >>= 1) ss += __shfl_xor(ss, off, 32);
    if (lane == 0) red[wave] = ss;
    __syncthreads();
    if (wave == 0) {
        float v = red[lane];
        for (int off = 16; off > 0; off >>= 1) v += __shfl_xor(v, off, 32);
        if (lane == 0) bcast[1] = v;
    }
    __syncthreads();
    if (tid < Q961) att2[(size_t)lb * Q961 + tid] = e / bcast[1] + 1.0f;
}

// ---------------------------------------------------------------------------
// Kernel 4/5: o = f * att (elementwise, fully coalesced)
// ---------------------------------------------------------------------------
__global__ void cam_scale(const float* __restrict__ f, const float* __restrict__ att,
                          float* __restrict__ out, int PQ, int total) {
    int i = blockIdx.x * blockDim.x + threadIdx.x;
    if (i >= total) return;
    int p  = i % PQ;
    int lb = i / (PQ * C);
    out[i] = f[i] * att[(size_t)lb * PQ + p];
}

// ---------------------------------------------------------------------------
extern "C" void kernel_launch(void* const* d_in, const int* in_sizes, int n_in,
                              void* d_out, int out_size, void* d_ws, size_t ws_size,
                              hipStream_t stream) {
    const float* zf  = (const float*)d_in[0];
    const float* xf  = (const float*)d_in[1];
    const float* W1  = (const float*)d_in[2];
    const float* b1  = (const float*)d_in[3];
    const float* g1  = (const float*)d_in[4];
    const float* be1 = (const float*)d_in[5];
    const float* rm1 = (const float*)d_in[6];
    const float* rv1 = (const float*)d_in[7];
    const float* W2  = (const float*)d_in[8];
    const float* b2  = (const float*)d_in[9];
    const float* W3  = (const float*)d_in[10];
    const float* b3  = (const float*)d_in[11];
    const float* g3  = (const float*)d_in[12];
    const float* be3 = (const float*)d_in[13];
    const float* rm3 = (const float*)d_in[14];
    const float* rv3 = (const float*)d_in[15];
    const float* W4  = (const float*)d_in[16];
    const float* b4  = (const float*)d_in[17];

    // workspace layout (floats), ~22 MB total
    float* ws      = (float*)d_ws;
    float* inv1    = ws;                                  // 96*49
    float* inv2    = inv1 + LB * P49;                     // 96*961
    float* A       = inv2 + LB * Q961;                    // 96*49*961
    float* att1    = A    + (size_t)LB * P49 * Q961;      // 96*49
    float* att2    = att1 + LB * P49;                     // 96*961
    float* partial = att2 + LB * Q961;                    // 8*96*961

    float* out1 = (float*)d_out;                          // 3*32*256*49
    float* out2 = out1 + (size_t)LB * C * P49;            // 3*32*256*961

    cam_norm<<<LB, 64, 0, stream>>>(zf, inv1, P49);
    cam_norm2_partial<<<dim3(LB, NCH), 128, 0, stream>>>(xf, partial);
    cam_norm2_finish<<<LB, 256, 0, stream>>>(partial, inv2);

    cam_gemm<<<dim3(LB, (Q961 + 15) / 16), 128, 0, stream>>>(zf, xf, inv1, inv2, A);

    cam_attn<<<LB, 1024, 0, stream>>>(A, W1, b1, g1, be1, rm1, rv1, W2, b2,
                                      W3, b3, g3, be3, rm3, rv3, W4, b4,
                                      att1, att2);

    int t1 = LB * C * P49;
    int t2 = LB * C * Q961;
    cam_scale<<<(t1 + 255) / 256, 256, 0, stream>>>(zf, att1, out1, P49,  t1);
    cam_scale<<<(t2 + 255) / 256, 256, 0, stream>>>(xf, att2, out2, Q961, t2);
}